// SpecNorm48_29557964931356
// MI455X (gfx1250) — compile-verified
//
#include <hip/hip_runtime.h>
#include <math.h>

// SpecNorm EMA-normalization via CDNA5 f32 WMMA.
// s_t = a*s_{t-1} + (1-a)*|x_t|  ==>  per 16-step chunk:
//   S = L @ XA + p * s_carry ,  L[i][j] = (1-a)*a^(i-j) (i>=j), p[i]=a^(i+1)
// computed with 4 chained V_WMMA_F32_16X16X4_F32 (K=16 total).

typedef float v2f __attribute__((ext_vector_type(2)));
typedef float v8f __attribute__((ext_vector_type(8)));

#define ALPHA_F      0.99f
#define ONE_M_ALPHA  0.01f
#define EPS_F        1e-12f
#define B_  32
#define T_  2000
#define F_  481
#define CHUNK 16
#define NCHUNK (T_ / CHUNK)   // 125, exact

__device__ __forceinline__ float alpha_pow(int e) {
    float r = 1.0f;
    #pragma unroll 1
    for (int i = 0; i < e; ++i) r *= ALPHA_F;
    return r;
}

__global__ __launch_bounds__(128) void
SpecNorm48_wmma_kernel(const float* __restrict__ x,
                       const float* __restrict__ s0,
                       float* __restrict__ out) {
    const int lane = threadIdx.x & 31;
    const int wid  = threadIdx.x >> 5;
    const int tile = blockIdx.x * 4 + wid;     // f-tile of 16 columns
    const int b    = blockIdx.y;
    const int f0   = tile * 16;
    if (f0 >= F_) return;                      // uniform per wave -> EXEC stays full

    const int  hi     = lane >> 4;             // 0: lanes 0-15, 1: lanes 16-31
    const int  n      = lane & 15;             // column (and A-row) index
    const int  fcol   = f0 + n;
    const bool fvalid = fcol < F_;
    const int  fL     = fvalid ? fcol : (F_ - 1);   // clamped: loads always in-bounds

    // ---- A operand: constant lower-triangular L, per ISA 16x4 f32 layout ----
    // lane holds row m=n; VGPR pair per K-slice k0: columns k = 4*k0 + 2*hi + {0,1}
    v2f Amat[4];
    #pragma unroll
    for (int k0 = 0; k0 < 4; ++k0) {
        const int k = 4 * k0 + 2 * hi;
        Amat[k0].x = (n >= k)     ? ONE_M_ALPHA * alpha_pow(n - k)     : 0.0f;
        Amat[k0].y = (n >= k + 1) ? ONE_M_ALPHA * alpha_pow(n - k - 1) : 0.0f;
    }
    // C/D rows held by this lane: row = r + 8*hi ; carry-in coefficient a^(row+1)
    float prow[8];
    #pragma unroll
    for (int r = 0; r < 8; ++r) prow[r] = alpha_pow(r + 8 * hi + 1);

    const long long basebt = (long long)b * T_;        // (b,t) row index base
    const float2* __restrict__ xp = (const float2*)x;  // one float2 per (b,t,f)
    float2* __restrict__ op = (float2*)out;

    float carry = s0[fL];   // reference initializes s with s0 (broadcast over B)

    for (int c = 0; c < NCHUNK; ++c) {
        const int tb = c * CHUNK;

        v8f acc;
        #pragma unroll
        for (int r = 0; r < 8; ++r) acc[r] = prow[r] * carry;

        #pragma unroll
        for (int k0 = 0; k0 < 4; ++k0) {
            const int k = 4 * k0 + 2 * hi;     // B rows this lane supplies
            // Branchless, clamped loads: invalid columns produce don't-care
            // values that are masked at the store, so EXEC stays full here.
            const float2 p0 = xp[(basebt + tb + k    ) * F_ + fL];
            const float2 p1 = xp[(basebt + tb + k + 1) * F_ + fL];
            if (c + 1 < NCHUNK) {              // next-chunk prefetch (always in-bounds)
                __builtin_prefetch(&xp[(basebt + tb + CHUNK + k) * F_ + fL], 0, 3);
            }
            v2f bmat;
            bmat.x = __builtin_amdgcn_sqrtf(p0.x * p0.x + p0.y * p0.y); // raw v_sqrt_f32
            bmat.y = __builtin_amdgcn_sqrtf(p1.x * p1.x + p1.y * p1.y);
            // D = A*B + C ; 8-arg form: (neg_a, A, neg_b, B, c_mod, C, reuse_a, reuse_b)
            acc = __builtin_amdgcn_wmma_f32_16x16x4_f32(false, Amat[k0], false, bmat,
                                                        (short)0, acc, false, false);
        }

        // carry for next chunk = S[15][n], held in VGPR7 of hi-half lanes;
        // broadcast to the column partner lane (n | 16) -> ds_bpermute path.
        carry = __shfl(acc[7], n | 16, 32);

        if (fvalid) {                          // uniform for 30 of 31 tiles
            #pragma unroll
            for (int r = 0; r < 8; ++r) {
                const int t = tb + r + 8 * hi;
                const long long idx = (basebt + t) * F_ + fcol;
                const float2 p = xp[idx];                         // L0-hot reload
                const float inv = __builtin_amdgcn_rsqf(acc[r] + EPS_F); // raw v_rsq_f32
                op[idx] = make_float2(p.x * inv, p.y * inv);
            }
        }
    }
}

extern "C" void kernel_launch(void* const* d_in, const int* in_sizes, int n_in,
                              void* d_out, int out_size, void* d_ws, size_t ws_size,
                              hipStream_t stream) {
    const float* x  = (const float*)d_in[0];   // [32, 2000, 481, 2] f32
    const float* s0 = (const float*)d_in[1];   // [481] f32
    float* out = (float*)d_out;                // [32, 2000, 481, 2] f32

    // 31 f-tiles of 16 columns (481 -> 31 tiles), 4 tiles per 128-thread block.
    dim3 grid((31 + 3) / 4, B_, 1);            // (8, 32)
    dim3 block(128, 1, 1);
    SpecNorm48_wmma_kernel<<<grid, block, 0, stream>>>(x, s0, out);
}